// EMMaskGenerator_73924977099073
// MI455X (gfx1250) — compile-verified
//
#include <hip/hip_runtime.h>
#include <math.h>

// Problem constants (from reference setup_inputs)
#define BB   16
#define SS   8
#define CC   3
#define CP   4          // C+1
#define KK   6
#define NPIX 16384      // 128*128
#define EPSF 1.19209290e-07f

typedef float v2f __attribute__((ext_vector_type(2)));
typedef float v8f __attribute__((ext_vector_type(8)));

// ---------------------------------------------------------------------------
// Phase A: one wave per block. WMMA f32 16x16x4 GEMM:
//   L (M=32 rows: 24 shifted(s,a) + 4 padded(b) + 4 zero)  x
//   R (N=32 cols: 24 gp(k)*padded(b) + 8 zero)   reduced over pixels (K dim).
// Result rows 0..23 -> ops_first, rows 24..27 -> ops_second (incl. gp_sum at
// [k,3,3]). Accumulated across pixel chunks, scattered with f32 atomics.
// ---------------------------------------------------------------------------
__device__ __forceinline__ void scatterA(int u, int row, int col, float val,
                                         float* opsF, float* opsS) {
    if (col < 24) {
        int k = col >> 2, b = col & 3;
        if (row < 24) {
            int s = row / 3, a = row - 3 * s;
            atomicAdd(&opsF[((u * SS + s) * KK + k) * 12 + a * 4 + b], val);
        } else if (row < 28) {
            int a = row - 24;
            atomicAdd(&opsS[(u * KK + k) * 16 + a * 4 + b], val);
        }
    }
}

__global__ void __launch_bounds__(32)
phaseA(const float* __restrict__ img, const float* __restrict__ shf,
       const float* __restrict__ pm, const float* __restrict__ gm,
       int use_gm, float* opsF, float* opsS) {
    const int CHUNK = 512;
    const int nch = NPIX / CHUNK;                 // 32 chunks per batch
    int u = blockIdx.x / nch;
    int base = (blockIdx.x % nch) * CHUNK;
    int lane = threadIdx.x;

    __shared__ float Lbuf[28][33];                // stride 33: conflict-free
    __shared__ float Rbuf[24][33];

    v8f acc00 = {}, acc01 = {}, acc10 = {}, acc11 = {};

    const int mrow = lane & 15;
    const int kk = (lane < 16) ? 0 : 2;           // K(pixel) split per ISA A/B layout
    const int r2 = 16 + mrow;

    for (int t = 0; t < CHUNK; t += 32) {
        int pix = base + t + lane;
        // padded
        float p0 = img[(u * CC + 0) * NPIX + pix];
        float p1 = img[(u * CC + 1) * NPIX + pix];
        float p2 = img[(u * CC + 2) * NPIX + pix];
        // softmax over K (+EPS)
        float lg[KK]; float mx = -1e30f;
#pragma unroll
        for (int k = 0; k < KK; k++) {
            float v = pm[(u * KK + k) * NPIX + pix];
            float g = gm[(u * KK + k) * NPIX + pix];
            v += use_gm ? g : 0.0f;
            lg[k] = v; mx = fmaxf(mx, v);
        }
        float se = 0.0f;
#pragma unroll
        for (int k = 0; k < KK; k++) { lg[k] = __expf(lg[k] - mx); se += lg[k]; }
        float inv = 1.0f / se;
        float gp[KK];
#pragma unroll
        for (int k = 0; k < KK; k++) gp[k] = lg[k] * inv + EPSF;

        // stage L and R into LDS (lane == pixel within tile)
#pragma unroll
        for (int r = 0; r < 24; r++)
            Lbuf[r][lane] = shf[(u * 24 + r) * NPIX + pix];
        Lbuf[24][lane] = p0; Lbuf[25][lane] = p1;
        Lbuf[26][lane] = p2; Lbuf[27][lane] = 1.0f;
        float pv[CP] = {p0, p1, p2, 1.0f};
#pragma unroll
        for (int k = 0; k < KK; k++)
#pragma unroll
            for (int b = 0; b < CP; b++)
                Rbuf[k * 4 + b][lane] = gp[k] * pv[b];
        __syncthreads();

        // 8 WMMA groups of 4 pixels each
#pragma unroll
        for (int g = 0; g < 8; g++) {
            int px = g * 4 + kk;
            v2f a0, a1, b0, b1;
            a0.x = Lbuf[mrow][px];  a0.y = Lbuf[mrow][px + 1];
            a1.x = (r2 < 28) ? Lbuf[r2][px]     : 0.0f;
            a1.y = (r2 < 28) ? Lbuf[r2][px + 1] : 0.0f;
            b0.x = Rbuf[mrow][px];  b0.y = Rbuf[mrow][px + 1];
            b1.x = (r2 < 24) ? Rbuf[r2][px]     : 0.0f;
            b1.y = (r2 < 24) ? Rbuf[r2][px + 1] : 0.0f;
            acc00 = __builtin_amdgcn_wmma_f32_16x16x4_f32(false, a0, false, b0,
                        (short)0, acc00, false, false);
            acc01 = __builtin_amdgcn_wmma_f32_16x16x4_f32(false, a0, false, b1,
                        (short)0, acc01, false, false);
            acc10 = __builtin_amdgcn_wmma_f32_16x16x4_f32(false, a1, false, b0,
                        (short)0, acc10, false, false);
            acc11 = __builtin_amdgcn_wmma_f32_16x16x4_f32(false, a1, false, b1,
                        (short)0, acc11, false, false);
        }
        __syncthreads();
    }

    // scatter C/D: VGPR v, lanes<16 -> M=v, lanes>=16 -> M=v+8; N=lane&15
    int n0 = lane & 15;
    int mh = (lane < 16) ? 0 : 8;
#pragma unroll
    for (int v = 0; v < 8; v++) {
        scatterA(u, v + mh,      n0,      acc00[v], opsF, opsS);
        scatterA(u, v + mh,      16 + n0, acc01[v], opsF, opsS);
        scatterA(u, 16 + v + mh, n0,      acc10[v], opsF, opsS);
        scatterA(u, 16 + v + mh, 16 + n0, acc11[v], opsF, opsS);
    }
}

// ---------------------------------------------------------------------------
// Phase B: per (u,k): norm=1/trace, invert 4x4 (ops_second*norm + EPS*I),
// then ops[u,s,k] = (ops_first*norm) @ inv  for all s.
// ---------------------------------------------------------------------------
__device__ inline void inv4(const float m[16], float out[16]) {
    float i[16];
    i[0]  =  m[5]*m[10]*m[15] - m[5]*m[11]*m[14] - m[9]*m[6]*m[15] + m[9]*m[7]*m[14] + m[13]*m[6]*m[11] - m[13]*m[7]*m[10];
    i[4]  = -m[4]*m[10]*m[15] + m[4]*m[11]*m[14] + m[8]*m[6]*m[15] - m[8]*m[7]*m[14] - m[12]*m[6]*m[11] + m[12]*m[7]*m[10];
    i[8]  =  m[4]*m[9]*m[15]  - m[4]*m[11]*m[13] - m[8]*m[5]*m[15] + m[8]*m[7]*m[13] + m[12]*m[5]*m[11] - m[12]*m[7]*m[9];
    i[12] = -m[4]*m[9]*m[14]  + m[4]*m[10]*m[13] + m[8]*m[5]*m[14] - m[8]*m[6]*m[13] - m[12]*m[5]*m[10] + m[12]*m[6]*m[9];
    i[1]  = -m[1]*m[10]*m[15] + m[1]*m[11]*m[14] + m[9]*m[2]*m[15] - m[9]*m[3]*m[14] - m[13]*m[2]*m[11] + m[13]*m[3]*m[10];
    i[5]  =  m[0]*m[10]*m[15] - m[0]*m[11]*m[14] - m[8]*m[2]*m[15] + m[8]*m[3]*m[14] + m[12]*m[2]*m[11] - m[12]*m[3]*m[10];
    i[9]  = -m[0]*m[9]*m[15]  + m[0]*m[11]*m[13] + m[8]*m[1]*m[15] - m[8]*m[3]*m[13] - m[12]*m[1]*m[11] + m[12]*m[3]*m[9];
    i[13] =  m[0]*m[9]*m[14]  - m[0]*m[10]*m[13] - m[8]*m[1]*m[14] + m[8]*m[2]*m[13] + m[12]*m[1]*m[10] - m[12]*m[2]*m[9];
    i[2]  =  m[1]*m[6]*m[15]  - m[1]*m[7]*m[14]  - m[5]*m[2]*m[15] + m[5]*m[3]*m[14] + m[13]*m[2]*m[7]  - m[13]*m[3]*m[6];
    i[6]  = -m[0]*m[6]*m[15]  + m[0]*m[7]*m[14]  + m[4]*m[2]*m[15] - m[4]*m[3]*m[14] - m[12]*m[2]*m[7]  + m[12]*m[3]*m[6];
    i[10] =  m[0]*m[5]*m[15]  - m[0]*m[7]*m[13]  - m[4]*m[1]*m[15] + m[4]*m[3]*m[13] + m[12]*m[1]*m[7]  - m[12]*m[3]*m[5];
    i[14] = -m[0]*m[5]*m[14]  + m[0]*m[6]*m[13]  + m[4]*m[1]*m[14] - m[4]*m[2]*m[13] - m[12]*m[1]*m[6]  + m[12]*m[2]*m[5];
    i[3]  = -m[1]*m[6]*m[11]  + m[1]*m[7]*m[10]  + m[5]*m[2]*m[11] - m[5]*m[3]*m[10] - m[9]*m[2]*m[7]   + m[9]*m[3]*m[6];
    i[7]  =  m[0]*m[6]*m[11]  - m[0]*m[7]*m[10]  - m[4]*m[2]*m[11] + m[4]*m[3]*m[10] + m[8]*m[2]*m[7]   - m[8]*m[3]*m[6];
    i[11] = -m[0]*m[5]*m[11]  + m[0]*m[7]*m[9]   + m[4]*m[1]*m[11] - m[4]*m[3]*m[9]  - m[8]*m[1]*m[7]   + m[8]*m[3]*m[5];
    i[15] =  m[0]*m[5]*m[10]  - m[0]*m[6]*m[9]   - m[4]*m[1]*m[10] + m[4]*m[2]*m[9]  + m[8]*m[1]*m[6]   - m[8]*m[2]*m[5];
    float det = m[0]*i[0] + m[1]*i[4] + m[2]*i[8] + m[3]*i[12];
    float id = 1.0f / det;
#pragma unroll
    for (int j = 0; j < 16; j++) out[j] = i[j] * id;
}

__global__ void phaseB(const float* __restrict__ opsF,
                       const float* __restrict__ opsS, float* ops) {
    int t = blockIdx.x * blockDim.x + threadIdx.x;
    if (t >= BB * KK) return;
    int u = t / KK, k = t % KK;
    const float* src = &opsS[(u * KK + k) * 16];
    float M[16];
    float tr = src[0] + src[5] + src[10] + src[15];
    float norm = 1.0f / tr;
#pragma unroll
    for (int j = 0; j < 16; j++) M[j] = src[j] * norm;
    M[0] += EPSF; M[5] += EPSF; M[10] += EPSF; M[15] += EPSF;
    float Mi[16];
    inv4(M, Mi);
    for (int s = 0; s < SS; s++) {
        const float* F = &opsF[((u * SS + s) * KK + k) * 12];
        float* O = &ops[((u * SS + s) * KK + k) * 12];
#pragma unroll
        for (int a = 0; a < CC; a++)
#pragma unroll
            for (int c = 0; c < CP; c++) {
                float acc = 0.0f;
#pragma unroll
                for (int b = 0; b < CP; b++) acc += F[a * 4 + b] * Mi[b * 4 + c];
                O[a * 4 + c] = norm * acc;
            }
    }
}

// ---------------------------------------------------------------------------
// Phase C: per (u,k) block: accumulate symmetric sigmas[s][3x3] (6 entries)
// weighted by pp = gp / gp_sum. Wave32 shuffle reduction + atomics.
// ---------------------------------------------------------------------------
__global__ void __launch_bounds__(256)
phaseC(const float* __restrict__ img, const float* __restrict__ shf,
       const float* __restrict__ pm, const float* __restrict__ gm,
       int use_gm, const float* __restrict__ ops,
       const float* __restrict__ opsS, float* sig) {
    const int CH = 4096;
    const int nch = NPIX / CH;                     // 4
    int uk = blockIdx.x / nch;
    int base = (blockIdx.x % nch) * CH;
    int u = uk / KK, k = uk % KK;

    __shared__ float sOps[SS * 12];
    __shared__ float sPn;
    if (threadIdx.x < SS * 12) {
        int s = threadIdx.x / 12, m = threadIdx.x % 12;
        sOps[threadIdx.x] = ops[((u * SS + s) * KK + k) * 12 + m];
    }
    if (threadIdx.x == 0) sPn = 1.0f / opsS[(u * KK + k) * 16 + 15];
    __syncthreads();

    float acc[SS * 6];
#pragma unroll
    for (int j = 0; j < SS * 6; j++) acc[j] = 0.0f;

    for (int i = 0; i < 16; i++) {
        int pix = base + i * 256 + threadIdx.x;
        float p0 = img[(u * CC + 0) * NPIX + pix];
        float p1 = img[(u * CC + 1) * NPIX + pix];
        float p2 = img[(u * CC + 2) * NPIX + pix];
        float lg[KK]; float mx = -1e30f;
#pragma unroll
        for (int kq = 0; kq < KK; kq++) {
            float v = pm[(u * KK + kq) * NPIX + pix];
            float g = gm[(u * KK + kq) * NPIX + pix];
            v += use_gm ? g : 0.0f;
            lg[kq] = v; mx = fmaxf(mx, v);
        }
        float se = 0.0f;
#pragma unroll
        for (int kq = 0; kq < KK; kq++) { lg[kq] = __expf(lg[kq] - mx); se += lg[kq]; }
        float gpk = lg[k] / se + EPSF;
        float pp = gpk * sPn;
        float sh[24];
#pragma unroll
        for (int j = 0; j < 24; j++) sh[j] = shf[(u * 24 + j) * NPIX + pix];
#pragma unroll
        for (int s = 0; s < SS; s++) {
            const float* O = &sOps[s * 12];
            float d0 = sh[s*3+0] - (O[0]*p0 + O[1]*p1 + O[2]*p2  + O[3]);
            float d1 = sh[s*3+1] - (O[4]*p0 + O[5]*p1 + O[6]*p2  + O[7]);
            float d2 = sh[s*3+2] - (O[8]*p0 + O[9]*p1 + O[10]*p2 + O[11]);
            acc[s*6+0] += pp * d0 * d0;
            acc[s*6+1] += pp * d0 * d1;
            acc[s*6+2] += pp * d0 * d2;
            acc[s*6+3] += pp * d1 * d1;
            acc[s*6+4] += pp * d1 * d2;
            acc[s*6+5] += pp * d2 * d2;
        }
    }
    // wave32 tree reduction, lane 0 of each wave commits with atomics
#pragma unroll
    for (int j = 0; j < SS * 6; j++) {
        float v = acc[j];
        for (int off = 16; off > 0; off >>= 1) v += __shfl_down(v, off, 32);
        if ((threadIdx.x & 31) == 0)
            atomicAdd(&sig[((u * SS + j / 6) * KK + k) * 6 + (j % 6)], v);
    }
}

// ---------------------------------------------------------------------------
// Phase D: per (u,s,k): add EPS*I, invert symmetric 3x3, logdet -> sum over s.
// ---------------------------------------------------------------------------
__global__ void phaseD(const float* __restrict__ sig, float* siginv,
                       float* logdet) {
    int t = blockIdx.x * blockDim.x + threadIdx.x;
    if (t >= BB * SS * KK) return;
    int u = t / (SS * KK), k = t % KK;
    const float* m = &sig[t * 6];
    float m0 = m[0] + EPSF, m1 = m[1], m2 = m[2];
    float m3 = m[3] + EPSF, m4 = m[4], m5 = m[5] + EPSF;
    float c00 = m3 * m5 - m4 * m4;
    float c01 = m2 * m4 - m1 * m5;
    float c02 = m1 * m4 - m2 * m3;
    float det = m0 * c00 + m1 * c01 + m2 * c02;
    float id = 1.0f / det;
    float* o = &siginv[t * 6];
    o[0] = c00 * id; o[1] = c01 * id; o[2] = c02 * id;
    o[3] = (m0 * m5 - m2 * m2) * id;
    o[4] = (m1 * m2 - m0 * m4) * id;
    o[5] = (m0 * m3 - m1 * m1) * id;
    atomicAdd(&logdet[u * KK + k], __logf(det));
}

// ---------------------------------------------------------------------------
// Phase E: per pixel: quad = sum_s diff . Sinv . diff per k; write masks.
// ---------------------------------------------------------------------------
__global__ void __launch_bounds__(256)
phaseE(const float* __restrict__ img, const float* __restrict__ shf,
       const float* __restrict__ ops, const float* __restrict__ siginv,
       const float* __restrict__ logdet, float* gm) {
    const int CH = 2048;
    int u = blockIdx.x / (NPIX / CH);
    int base = (blockIdx.x % (NPIX / CH)) * CH;

    __shared__ float sOps[SS * KK * 12];   // 576, contiguous for fixed u
    __shared__ float sSi[SS * KK * 6];     // 288
    __shared__ float sLd[KK];
    for (int i = threadIdx.x; i < SS * KK * 12; i += 256)
        sOps[i] = ops[u * SS * KK * 12 + i];
    for (int i = threadIdx.x; i < SS * KK * 6; i += 256)
        sSi[i] = siginv[u * SS * KK * 6 + i];
    if (threadIdx.x < KK) sLd[threadIdx.x] = logdet[u * KK + threadIdx.x];
    __syncthreads();

    for (int i = 0; i < 8; i++) {
        int pix = base + i * 256 + threadIdx.x;
        float p0 = img[(u * CC + 0) * NPIX + pix];
        float p1 = img[(u * CC + 1) * NPIX + pix];
        float p2 = img[(u * CC + 2) * NPIX + pix];
        float sh[24];
#pragma unroll
        for (int j = 0; j < 24; j++) sh[j] = shf[(u * 24 + j) * NPIX + pix];
#pragma unroll
        for (int k = 0; k < KK; k++) {
            float quad = 0.0f;
#pragma unroll
            for (int s = 0; s < SS; s++) {
                const float* O = &sOps[(s * KK + k) * 12];
                float d0 = sh[s*3+0] - (O[0]*p0 + O[1]*p1 + O[2]*p2  + O[3]);
                float d1 = sh[s*3+1] - (O[4]*p0 + O[5]*p1 + O[6]*p2  + O[7]);
                float d2 = sh[s*3+2] - (O[8]*p0 + O[9]*p1 + O[10]*p2 + O[11]);
                const float* S = &sSi[(s * KK + k) * 6];
                quad += S[0]*d0*d0 + S[3]*d1*d1 + S[5]*d2*d2
                      + 2.0f * (S[1]*d0*d1 + S[2]*d0*d2 + S[4]*d1*d2);
            }
            gm[(u * KK + k) * NPIX + pix] = -0.5f * quad - 0.5f * sLd[k];
        }
    }
}

// ---------------------------------------------------------------------------
extern "C" void kernel_launch(void* const* d_in, const int* in_sizes, int n_in,
                              void* d_out, int out_size, void* d_ws,
                              size_t ws_size, hipStream_t stream) {
    const float* img = (const float*)d_in[0];   // [16,3,128,128]
    const float* shf = (const float*)d_in[1];   // [16,8,3,128,128]
    const float* pm  = (const float*)d_in[2];   // [16,6,128,128]
    float* gm = (float*)d_out;                  // [16,6,128,128] = state + out

    float* ws   = (float*)d_ws;
    float* opsF = ws;                 // 16*8*6*12 = 9216
    float* opsS = opsF + 9216;        // 16*6*16   = 1536
    float* sig  = opsS + 1536;        // 16*8*6*6  = 4608
    float* logd = sig + 4608;         // 16*6      = 96
    float* ops  = logd + 96;          // 9216
    float* sinv = ops + 9216;         // 4608
    size_t zbytes = (size_t)(9216 + 1536 + 4608 + 96) * sizeof(float);

    for (int step = 0; step < 3; step++) {
        int use_gm = (step > 0) ? 1 : 0;
        hipMemsetAsync(ws, 0, zbytes, stream);
        phaseA<<<dim3(BB * (NPIX / 512)), dim3(32), 0, stream>>>(
            img, shf, pm, gm, use_gm, opsF, opsS);
        phaseB<<<dim3(1), dim3(128), 0, stream>>>(opsF, opsS, ops);
        phaseC<<<dim3(BB * KK * (NPIX / 4096)), dim3(256), 0, stream>>>(
            img, shf, pm, gm, use_gm, ops, opsS, sig);
        phaseD<<<dim3(3), dim3(256), 0, stream>>>(sig, sinv, logd);
        phaseE<<<dim3(BB * (NPIX / 2048)), dim3(256), 0, stream>>>(
            img, shf, ops, sinv, logd, gm);
    }
}